// UVTo3D_78984448573770
// MI455X (gfx1250) — compile-verified
//
#include <hip/hip_runtime.h>
#include <cstdint>

#define PI_F      3.14159265358979f
#define TWO_PI_F  6.28318530717959f

// ---------------- CDNA5 async global->LDS helpers ----------------
#if defined(__HIP_DEVICE_COMPILE__)
#define LDS_AS __attribute__((address_space(3)))
#define GLB_AS __attribute__((address_space(1)))
typedef int v4i_t __attribute__((vector_size(4 * sizeof(int))));
#endif

__device__ __forceinline__ void async_g2l_b32(const void* g, void* l) {
#if defined(__HIP_DEVICE_COMPILE__)
#if __has_builtin(__builtin_amdgcn_global_load_async_to_lds_b32)
    __builtin_amdgcn_global_load_async_to_lds_b32(
        (GLB_AS int*)(g), (LDS_AS int*)(l), 0, 0);
#else
    asm volatile("global_load_async_to_lds_b32 %0, %1, off"
                 :
                 : "v"((unsigned int)(unsigned long long)(LDS_AS void*)l),
                   "v"((unsigned long long)g)
                 : "memory");
#endif
#else
    (void)g; (void)l;
#endif
}

__device__ __forceinline__ void async_g2l_b128(const void* g, void* l) {
#if defined(__HIP_DEVICE_COMPILE__)
#if __has_builtin(__builtin_amdgcn_global_load_async_to_lds_b128)
    __builtin_amdgcn_global_load_async_to_lds_b128(
        (GLB_AS v4i_t*)(g), (LDS_AS v4i_t*)(l), 0, 0);
#else
    asm volatile("global_load_async_to_lds_b128 %0, %1, off"
                 :
                 : "v"((unsigned int)(unsigned long long)(LDS_AS void*)l),
                   "v"((unsigned long long)g)
                 : "memory");
#endif
#else
    (void)g; (void)l;
#endif
}

__device__ __forceinline__ void async_wait_all() {
#if defined(__HIP_DEVICE_COMPILE__)
#if __has_builtin(__builtin_amdgcn_s_wait_asynccnt)
    __builtin_amdgcn_s_wait_asynccnt(0);
#else
    asm volatile("s_wait_asynccnt 0" ::: "memory");
#endif
#endif
}

// Cooperative staging of n dwords from global to LDS (async DMA path).
__device__ __forceinline__ void stage_to_lds(const float* __restrict__ g,
                                             float* l, int n) {
    const int t = threadIdx.x, bs = blockDim.x;
    const int n4 = n & ~3;
    for (int i = 4 * t; i < n4; i += 4 * bs)
        async_g2l_b128(g + i, l + i);
    for (int i = n4 + t; i < n; i += bs)
        async_g2l_b32(g + i, l + i);
}

// ---------------- math helpers ----------------
struct F3 { float x, y, z; };

__device__ __forceinline__ F3 f3sub(F3 a, F3 b) {
    return {a.x - b.x, a.y - b.y, a.z - b.z};
}
__device__ __forceinline__ F3 f3cross(F3 a, F3 b) {
    return {a.y * b.z - a.z * b.y,
            a.z * b.x - a.x * b.z,
            a.x * b.y - a.y * b.x};
}
__device__ __forceinline__ float f3norm(F3 a) {
    return sqrtf(a.x * a.x + a.y * a.y + a.z * a.z);
}
// phi = 2*pi*(u-0.5), theta = pi*(v-0.5); args in [-pi,pi] so v_sin/v_cos are exact enough
__device__ __forceinline__ F3 uv_to_sphere(float u, float v) {
    float sp, cp, st, ct;
    __sincosf(TWO_PI_F * (u - 0.5f), &sp, &cp);
    __sincosf(PI_F * (v - 0.5f), &st, &ct);
    return {ct * cp, ct * sp, st};
}

// ---------------- kernel ----------------
__global__ void __launch_bounds__(256)
UVTo3D_kernel(const float* __restrict__ uv,
              const float* __restrict__ verts,
              const float* __restrict__ uv_verts,
              const int* __restrict__ faces,
              const int* __restrict__ face_inds,
              const int* __restrict__ pH,
              const int* __restrict__ pW,
              float* __restrict__ out,
              int N, int V, int F, int off_uvv, int off_faces)
{
    extern __shared__ float smem[];
    float* s_verts = smem;                    // 3*V floats
    float* s_uvv   = smem + off_uvv;          // 2*V floats
    int*   s_faces = (int*)(smem + off_faces);// 3*F ints

    // Async-DMA the small gather tables into LDS (28 KB << 320 KB/WGP).
    stage_to_lds(verts, s_verts, 3 * V);
    stage_to_lds(uv_verts, s_uvv, 2 * V);
    stage_to_lds((const float*)faces, (float*)s_faces, 3 * F);
    async_wait_all();      // this wave's async copies landed in LDS
    __syncthreads();       // all waves' copies landed

    const int W = pW[0];
    const int H = pH[0];
    const float sx = (float)(W - 1);
    const float sy = (float)(H - 1);

    const int stride = (int)(gridDim.x * blockDim.x);
    for (int i = (int)(blockIdx.x * blockDim.x + threadIdx.x); i < N; i += stride) {
        const float2 p2 = reinterpret_cast<const float2*>(uv)[i];
        // jnp.round == round-half-even == v_rndne_f32
        const int ix = (int)rintf(sx * p2.x);
        const int iy = (int)rintf(sy * p2.y);
        const int f  = face_inds[iy * W + ix];   // 4 MB table: L2-resident gather

        const int i0 = s_faces[3 * f + 0];
        const int i1 = s_faces[3 * f + 1];
        const int i2 = s_faces[3 * f + 2];

        const F3 vA = uv_to_sphere(s_uvv[2 * i0], s_uvv[2 * i0 + 1]);
        const F3 vB = uv_to_sphere(s_uvv[2 * i1], s_uvv[2 * i1 + 1]);
        const F3 vC = uv_to_sphere(s_uvv[2 * i2], s_uvv[2 * i2 + 1]);
        const F3 P  = uv_to_sphere(p2.x, p2.y);

        const F3 AB = f3sub(vB, vA);
        const F3 AC = f3sub(vC, vA);
        const F3 BC = f3sub(vC, vB);
        const F3 AP = f3sub(P, vA);
        const F3 BP = f3sub(P, vB);

        const float nBAC = f3norm(f3cross(AB, AC));
        const float w_ = f3norm(f3cross(AB, AP)) / nBAC;
        const float v_ = f3norm(f3cross(AC, AP)) / nBAC;
        const float u_ = f3norm(f3cross(BC, BP)) / nBAC;

        const float denom = fmaxf(fabsf(u_) + fabsf(v_) + fabsf(w_), 1e-12f);
        const float s = 1.0f / denom;
        const float bu = u_ * s, bv = v_ * s, bw = w_ * s;

        out[3 * i + 0] = bu * s_verts[3 * i0 + 0] + bv * s_verts[3 * i1 + 0] + bw * s_verts[3 * i2 + 0];
        out[3 * i + 1] = bu * s_verts[3 * i0 + 1] + bv * s_verts[3 * i1 + 1] + bw * s_verts[3 * i2 + 1];
        out[3 * i + 2] = bu * s_verts[3 * i0 + 2] + bv * s_verts[3 * i1 + 2] + bw * s_verts[3 * i2 + 2];
    }
}

// ---------------- launcher ----------------
extern "C" void kernel_launch(void* const* d_in, const int* in_sizes, int n_in,
                              void* d_out, int out_size, void* d_ws, size_t ws_size,
                              hipStream_t stream) {
    (void)n_in; (void)d_ws; (void)ws_size; (void)out_size;

    const float* uv        = (const float*)d_in[0];   // (N,2) f32
    const float* verts     = (const float*)d_in[1];   // (V,3) f32
    const float* uv_verts  = (const float*)d_in[2];   // (V,2) f32
    const int*   faces     = (const int*)d_in[3];     // (F,3) i32
    const int*   face_inds = (const int*)d_in[4];     // (H,W) i32
    const int*   pH        = (const int*)d_in[5];     // scalar
    const int*   pW        = (const int*)d_in[6];     // scalar
    float*       out       = (float*)d_out;           // (N,3) f32

    const int N = in_sizes[0] / 2;
    const int V = in_sizes[1] / 3;
    const int F = in_sizes[3] / 3;

    // LDS layout (16B-aligned sections for b128 async copies)
    auto align4 = [](int x) { return (x + 3) & ~3; };
    const int off_uvv   = align4(3 * V);
    const int off_faces = off_uvv + align4(2 * V);
    const int lds_dwords = off_faces + align4(3 * F);
    const size_t lds_bytes = (size_t)lds_dwords * 4;

    const int threads = 256;          // 8 wave32s per workgroup
    const int ppt = 8;                // points per thread (amortize LDS staging)
    long denom = (long)threads * ppt;
    int blocks = (int)((N + denom - 1) / denom);
    if (blocks < 1) blocks = 1;

    UVTo3D_kernel<<<blocks, threads, lds_bytes, stream>>>(
        uv, verts, uv_verts, faces, face_inds, pH, pW, out,
        N, V, F, off_uvv, off_faces);
}